// CovariantAttention_87806311400098
// MI455X (gfx1250) — compile-verified
//
#include <hip/hip_runtime.h>
#include <math.h>

// ---------------- problem constants ----------------
#define B_  4
#define N_  256
#define D_  256
#define DL_ 64
#define JD_ (D_ * DL_)   // 16384
#define G_S 1.0f
#define G_2 0.5f
#define G_1 0.3f

typedef float v2f __attribute__((ext_vector_type(2)));
typedef float v8f __attribute__((ext_vector_type(8)));

// workspace layout (in floats)
#define Q_OFF      0          // B*D        = 1024
#define K_OFF      1024       // B*N*D      = 262144
#define V_OFF      263168     // B*N*D      = 262144
#define DELTA_OFF  525312     // B*N*DL     = 65536
#define M_OFF      590848     // B*JD       = 65536
#define GB_OFF     656384     // B*N*DL     = 65536
#define AD_OFF     721920     // B*N*DL     = 65536
#define P_OFF      787456     // B*JD       = 65536
#define CP_OFF     852992     // 64*B*D     = 65536
#define LAM_OFF    918528     // B
#define TAU_OFF    918532     // B

static __device__ __forceinline__ v8f wmma4(v2f a, v2f b, v8f c) {
  return __builtin_amdgcn_wmma_f32_16x16x4_f32(false, a, false, b, (short)0, c,
                                               false, false);
}

// ---------------- q = Wx x + Wz z + Wv v + biases + bilinears ----------------
__global__ void k_build_q(const float* __restrict__ zq, const float* __restrict__ xq,
                          const float* __restrict__ vq, const float* __restrict__ vqg,
                          const float* __restrict__ Wx, const float* __restrict__ bx,
                          const float* __restrict__ Wz, const float* __restrict__ bz,
                          const float* __restrict__ Wv, const float* __restrict__ bv,
                          const float* __restrict__ wg, const float* __restrict__ wzv,
                          float* __restrict__ q_out) {
  int b = blockIdx.x, o = threadIdx.x;
  const float* x  = xq  + b * D_;
  const float* z  = zq  + b * DL_;
  const float* vv = vq  + b * D_;
  const float* vg = vqg + b * DL_;
  float acc = bx[o] + bz[o] + bv[o];
  for (int j = 0; j < D_;  ++j) acc += Wx[o * D_  + j] * x[j];
  for (int j = 0; j < DL_; ++j) acc += Wz[o * DL_ + j] * z[j];
  for (int j = 0; j < D_;  ++j) acc += Wv[o * D_  + j] * vv[j];
  const float* g  = wg  + o * (DL_ * DL_);
  const float* zz = wzv + o * (DL_ * DL_);
  for (int i = 0; i < DL_; ++i) {
    float zi = z[i], s = 0.f;
    for (int j = 0; j < DL_; ++j)
      s += g[i * DL_ + j] * z[j] + zz[i * DL_ + j] * vg[j];
    acc += zi * s;
  }
  q_out[b * D_ + o] = acc;
}

// ---------------- delta = zq - zk ; lambda, tau per batch ----------------
__global__ void k_build_delta(const float* __restrict__ zq, const float* __restrict__ zk,
                              float* __restrict__ delta, float* __restrict__ lam_ws,
                              float* __restrict__ tau_ws) {
  int b = blockIdx.x, n = threadIdx.x;
  const float* z = zq + b * DL_;
  const float* kzn = zk + (b * N_ + n) * DL_;
  float* dn = delta + (b * N_ + n) * DL_;
  for (int d = 0; d < DL_; ++d) dn[d] = z[d] - kzn[d];
  if (n == 0) {
    float ns = 0.f;
    for (int d = 0; d < DL_; ++d) ns += z[d] * z[d];
    float lam = 2.0f / (1.0f - ns + 1e-6f);
    lam_ws[b] = lam;
    tau_ws[b] = 16.0f / lam;  // sqrt(256)
  }
}

// ---------------- C[m,o] = X[m,:]·W[o,:] + bias[o]  (WMMA, fp32) ----------------
__global__ void k_proj(const float* __restrict__ X, const float* __restrict__ W,
                       const float* __restrict__ bias, float* __restrict__ C) {
  int wave = blockIdx.x * (blockDim.x >> 5) + (threadIdx.x >> 5);
  int lane = threadIdx.x & 31;
  int tm = wave >> 4, tn = wave & 15;       // 64 x 16 tiles
  int r = lane & 15, h = lane >> 4;
  const float* arow = X + (tm * 16 + r) * D_;
  const float* brow = W + (tn * 16 + r) * D_;
  v8f acc = {0.f, 0.f, 0.f, 0.f, 0.f, 0.f, 0.f, 0.f};
  for (int kk = 0; kk < D_; kk += 4) {
    v2f a = *(const v2f*)(arow + kk + 2 * h);
    v2f b = *(const v2f*)(brow + kk + 2 * h);
    acc = wmma4(a, b, acc);
  }
  int o = tn * 16 + r;
  float bo = bias[o];
#pragma unroll
  for (int v = 0; v < 8; ++v) {
    int m = tm * 16 + v + 8 * h;
    C[m * D_ + o] = acc[v] + bo;
  }
}

// ---------------- M[b, jd] = sum_i theta_comb[i, jd] * q[b, i] ----------------
// HBM-bound theta stream #1: per-lane speculative prefetch one K-block ahead.
__global__ void k_build_M(const float* __restrict__ q, const float* __restrict__ tb,
                          const float* __restrict__ te, const float* __restrict__ to_,
                          float* __restrict__ M) {
  int wave = blockIdx.x * (blockDim.x >> 5) + (threadIdx.x >> 5);
  int lane = threadIdx.x & 31;
  int r = lane & 15, h = lane >> 4;
  int ncol = wave * 16 + r;                 // jd column
  const float* arow = q + (r < 4 ? r : 0) * D_;
  float msk = (r < 4) ? 1.f : 0.f;
  v8f acc = {0.f, 0.f, 0.f, 0.f, 0.f, 0.f, 0.f, 0.f};
  for (int kk = 0; kk < D_; kk += 8) {
    // prefetch the theta rows of the next K-block (speculative; OOB dropped)
    int pf = (kk + 8 + 2 * h) * JD_ + ncol;
    __builtin_prefetch(tb + pf, 0, 1);
    __builtin_prefetch(te + pf, 0, 1);
    __builtin_prefetch(to_ + pf, 0, 1);
#pragma unroll
    for (int u = 0; u < 8; u += 4) {
      int k0 = kk + u + 2 * h;
      v2f a; a.x = arow[k0] * msk; a.y = arow[k0 + 1] * msk;
      int i0 = k0 * JD_ + ncol, i1 = i0 + JD_;
      v2f bfr;
      bfr.x = G_S * tb[i0] + G_2 * te[i0] + G_1 * to_[i0];
      bfr.y = G_S * tb[i1] + G_2 * te[i1] + G_1 * to_[i1];
      acc = wmma4(a, bfr, acc);
    }
  }
#pragma unroll
  for (int v = 0; v < 8; ++v) {
    int brow = v + 8 * h;
    if (brow < 4) M[brow * JD_ + ncol] = acc[v];
  }
}

// ---------------- G[b,n,d] = sum_j k[b,n,j] * M[b, j*DL + d] ----------------
__global__ void k_build_G(const float* __restrict__ kb, const float* __restrict__ M,
                          float* __restrict__ G) {
  int wave = blockIdx.x * (blockDim.x >> 5) + (threadIdx.x >> 5);
  int lane = threadIdx.x & 31;
  int b = wave >> 6, t = wave & 63;
  int tn = t >> 2, td = t & 3;
  int r = lane & 15, h = lane >> 4;
  const float* arow = kb + (b * N_ + tn * 16 + r) * D_;
  const float* Mb = M + b * JD_;
  int dcol = td * 16 + r;
  v8f acc = {0.f, 0.f, 0.f, 0.f, 0.f, 0.f, 0.f, 0.f};
  for (int kk = 0; kk < D_; kk += 4) {
    int k0 = kk + 2 * h;
    v2f a = *(const v2f*)(arow + k0);
    v2f bfr; bfr.x = Mb[k0 * DL_ + dcol]; bfr.y = Mb[(k0 + 1) * DL_ + dcol];
    acc = wmma4(a, bfr, acc);
  }
#pragma unroll
  for (int v = 0; v < 8; ++v) {
    int n = tn * 16 + v + 8 * h;
    G[(b * N_ + n) * DL_ + dcol] = acc[v];
  }
}

// ------- scores, softmax, screening, renorm; emit attention and a*delta -------
__global__ void k_attn(const float* __restrict__ q, const float* __restrict__ kb,
                       const float* __restrict__ delta, const float* __restrict__ G,
                       const float* __restrict__ tau_ws, const float* __restrict__ lam_ws,
                       const int* __restrict__ level, float* __restrict__ attn,
                       float* __restrict__ AD) {
  __shared__ float red[N_];
  int b = blockIdx.x, n = threadIdx.x;
  const float* qb = q + b * D_;
  const float* kn = kb + (b * N_ + n) * D_;
  float s = 0.f;
  for (int i = 0; i < D_; ++i) s += qb[i] * kn[i];
  const float* dn = delta + (b * N_ + n) * DL_;
  const float* gn = G + (b * N_ + n) * DL_;
  float corr = 0.f, dsq = 0.f;
  for (int d = 0; d < DL_; ++d) { corr += dn[d] * gn[d]; dsq += dn[d] * dn[d]; }
  s = (s - corr) / (tau_ws[b] + 1e-6f);
  // softmax
  red[n] = s; __syncthreads();
  for (int off = 128; off > 0; off >>= 1) {
    if (n < off) red[n] = fmaxf(red[n], red[n + off]);
    __syncthreads();
  }
  float mx = red[0]; __syncthreads();
  float e = __expf(s - mx);
  red[n] = e; __syncthreads();
  for (int off = 128; off > 0; off >>= 1) {
    if (n < off) red[n] += red[n + off];
    __syncthreads();
  }
  float a = e / red[0]; __syncthreads();
  // area-law screening
  float lam = lam_ws[b];
  float sigma = G_S * __expf(-(float)level[0]);
  a *= __expf(-sigma * 0.5f * lam * lam * dsq);
  red[n] = a; __syncthreads();
  for (int off = 128; off > 0; off >>= 1) {
    if (n < off) red[n] += red[n + off];
    __syncthreads();
  }
  a = a / (red[0] + 1e-6f);
  attn[b * N_ + n] = a;
  float* adn = AD + (b * N_ + n) * DL_;
  for (int d = 0; d < DL_; ++d) adn[d] = a * dn[d];
}

// ---------------- P[b, j*DL+d] = sum_n v[b,n,j] * AD[b,n,d] ----------------
__global__ void k_build_P(const float* __restrict__ vb, const float* __restrict__ AD,
                          float* __restrict__ P) {
  int wave = blockIdx.x * (blockDim.x >> 5) + (threadIdx.x >> 5);
  int lane = threadIdx.x & 31;
  int b = wave >> 6, t = wave & 63;
  int tj = t >> 2, td = t & 3;
  int r = lane & 15, h = lane >> 4;
  const float* vbB = vb + b * N_ * D_;
  const float* ADb = AD + b * N_ * DL_;
  int jrow = tj * 16 + r, dcol = td * 16 + r;
  v8f acc = {0.f, 0.f, 0.f, 0.f, 0.f, 0.f, 0.f, 0.f};
  for (int kk = 0; kk < N_; kk += 4) {
    int k0 = kk + 2 * h;
    v2f a;   a.x = vbB[k0 * D_ + jrow];   a.y = vbB[(k0 + 1) * D_ + jrow];
    v2f bfr; bfr.x = ADb[k0 * DL_ + dcol]; bfr.y = ADb[(k0 + 1) * DL_ + dcol];
    acc = wmma4(a, bfr, acc);
  }
#pragma unroll
  for (int v = 0; v < 8; ++v) {
    int j = tj * 16 + v + 8 * h;
    P[b * JD_ + j * DL_ + dcol] = acc[v];
  }
}

// ------- c_part[s][b][i] = sum_{jd in chunk s} theta_comb[i, jd] * P[b, jd] -------
// HBM-bound theta stream #2: per-lane row streams; prefetch one block ahead.
__global__ void k_corr(const float* __restrict__ P, const float* __restrict__ tb,
                       const float* __restrict__ te, const float* __restrict__ to_,
                       float* __restrict__ cpart) {
  int wave = blockIdx.x * (blockDim.x >> 5) + (threadIdx.x >> 5);
  int lane = threadIdx.x & 31;
  int ti = wave & 15, s = wave >> 4;  // 16 i-tiles, 64 K-chunks
  int r = lane & 15, h = lane >> 4;
  int kbase = s * 256;
  const float* arow = P + (r < 4 ? r : 0) * JD_ + kbase;
  float msk = (r < 4) ? 1.f : 0.f;
  int trow = (ti * 16 + r) * JD_ + kbase;
  v8f acc = {0.f, 0.f, 0.f, 0.f, 0.f, 0.f, 0.f, 0.f};
  for (int kk = 0; kk < 256; kk += 32) {
    // prefetch next 128B block of each per-lane theta row (speculative)
    int pf = trow + kk + 32 + 2 * h;
    __builtin_prefetch(tb + pf, 0, 1);
    __builtin_prefetch(te + pf, 0, 1);
    __builtin_prefetch(to_ + pf, 0, 1);
#pragma unroll
    for (int u = 0; u < 32; u += 4) {
      int k0 = kk + u + 2 * h;
      v2f a = *(const v2f*)(arow + k0);
      a.x *= msk; a.y *= msk;
      v2f t0 = *(const v2f*)(tb + trow + k0);
      v2f t1 = *(const v2f*)(te + trow + k0);
      v2f t2 = *(const v2f*)(to_ + trow + k0);
      v2f bfr;
      bfr.x = G_S * t0.x + G_2 * t1.x + G_1 * t2.x;
      bfr.y = G_S * t0.y + G_2 * t1.y + G_1 * t2.y;
      acc = wmma4(a, bfr, acc);
    }
  }
  int icol = ti * 16 + r;
#pragma unroll
  for (int v = 0; v < 8; ++v) {
    int brow = v + 8 * h;
    if (brow < 4) cpart[(s * B_ + brow) * D_ + icol] = acc[v];
  }
}

// ---------------- out[b,i] = sum_n attn*v - sum_s cpart ----------------
__global__ void k_final(const float* __restrict__ attn, const float* __restrict__ vb,
                        const float* __restrict__ cpart, float* __restrict__ out) {
  int b = blockIdx.x, i = threadIdx.x;
  const float* an = attn + b * N_;
  const float* vbB = vb + b * N_ * D_;
  float base = 0.f;
  for (int n = 0; n < N_; ++n) base += an[n] * vbB[n * D_ + i];
  float c = 0.f;
  for (int s = 0; s < 64; ++s) c += cpart[(s * B_ + b) * D_ + i];
  out[b * D_ + i] = base - c;
}

extern "C" void kernel_launch(void* const* d_in, const int* in_sizes, int n_in,
                              void* d_out, int out_size, void* d_ws, size_t ws_size,
                              hipStream_t stream) {
  const float* zq   = (const float*)d_in[0];
  const float* zk   = (const float*)d_in[1];
  const float* xq   = (const float*)d_in[2];
  const float* xk   = (const float*)d_in[3];
  const float* xv   = (const float*)d_in[4];
  const float* vq   = (const float*)d_in[5];
  const float* vqg  = (const float*)d_in[6];
  const float* Wx   = (const float*)d_in[7];
  const float* bx   = (const float*)d_in[8];
  const float* Wz   = (const float*)d_in[9];
  const float* bz   = (const float*)d_in[10];
  const float* Wv   = (const float*)d_in[11];
  const float* bv   = (const float*)d_in[12];
  const float* wg   = (const float*)d_in[13];
  const float* wzv  = (const float*)d_in[14];
  const float* Wk   = (const float*)d_in[15];
  const float* bk   = (const float*)d_in[16];
  const float* Wvl  = (const float*)d_in[17];
  const float* bvl  = (const float*)d_in[18];
  const float* tb   = (const float*)d_in[19];
  const float* te   = (const float*)d_in[20];
  const float* to_  = (const float*)d_in[21];
  const int*   lvl  = (const int*)d_in[22];

  float* ws    = (float*)d_ws;
  float* q     = ws + Q_OFF;
  float* kbuf  = ws + K_OFF;
  float* vbuf  = ws + V_OFF;
  float* delta = ws + DELTA_OFF;
  float* Mbuf  = ws + M_OFF;
  float* Gbuf  = ws + GB_OFF;
  float* AD    = ws + AD_OFF;
  float* Pbuf  = ws + P_OFF;
  float* cpart = ws + CP_OFF;
  float* lam   = ws + LAM_OFF;
  float* tau   = ws + TAU_OFF;

  float* out  = (float*)d_out;       // [B,D]
  float* attn = out + B_ * D_;       // [B,N]

  k_build_q<<<dim3(B_), dim3(D_), 0, stream>>>(zq, xq, vq, vqg, Wx, bx, Wz, bz,
                                               Wv, bv, wg, wzv, q);
  k_build_delta<<<dim3(B_), dim3(N_), 0, stream>>>(zq, zk, delta, lam, tau);
  k_proj<<<dim3(256), dim3(128), 0, stream>>>(xk, Wk, bk, kbuf);   // 1024 waves
  k_proj<<<dim3(256), dim3(128), 0, stream>>>(xv, Wvl, bvl, vbuf);
  k_build_M<<<dim3(256), dim3(128), 0, stream>>>(q, tb, te, to_, Mbuf);
  k_build_G<<<dim3(64), dim3(128), 0, stream>>>(kbuf, Mbuf, Gbuf);
  k_attn<<<dim3(B_), dim3(N_), 0, stream>>>(q, kbuf, delta, Gbuf, tau, lam, lvl,
                                            attn, AD);
  k_build_P<<<dim3(64), dim3(128), 0, stream>>>(vbuf, AD, Pbuf);
  k_corr<<<dim3(256), dim3(128), 0, stream>>>(Pbuf, tb, te, to_, cpart);
  k_final<<<dim3(B_), dim3(D_), 0, stream>>>(attn, vbuf, cpart, out);
}